// SchNetBlock_730144440877
// MI455X (gfx1250) — compile-verified
//
#include <hip/hip_runtime.h>
#include <hip/hip_bf16.h>
#include <math.h>

typedef __attribute__((ext_vector_type(2))) float v2f;
typedef __attribute__((ext_vector_type(8))) float v8f;

#define LN2F 0.69314718055994530942f
#define HSTRIDE 132   // 16x128 LDS tile row stride (floats): 4l+2h bank spread, b64-clean

__device__ __forceinline__ float ssp_f(float z) {
    // shifted softplus: log(1+e^z) - ln2, numerically stable
    float az = __builtin_fabsf(z);
    float e  = __expf(-az);
    return fmaxf(z, 0.0f) + __logf(1.0f + e) - LN2F;
}

// Async global->LDS copy (CDNA5): per-lane global addr -> per-lane LDS byte offset.
// Tracked by ASYNCcnt; drain with s_wait_asynccnt before LDS reads.
__device__ __forceinline__ void async_g2l_b32(unsigned lds_byte_off, const float* gptr) {
    asm volatile("global_load_async_to_lds_b32 %0, %1, off"
                 :: "v"(lds_byte_off),
                    "v"((unsigned long long)(uintptr_t)gptr)
                 : "memory");
}
__device__ __forceinline__ void async_drain() {
    asm volatile("s_wait_asynccnt 0x0" ::: "memory");
}

// Stage a KxN (N=128) weight matrix into LDS pre-swizzled into WMMA B-fragment
// order: fragment (ks,n) occupies 64 consecutive floats; lane reads floats
// [lane*2, lane*2+1] = W[4ks+2h + {0,1}][n*16+l]  ->  one conflict-free ds_load_b64.
__device__ __forceinline__ void fill_sw(unsigned sw_float_off, const float* __restrict__ W,
                                        int ksteps, int tid) {
    int nslots = ksteps * 256;                 // ksteps * 8 ntiles * 32 lanes
    for (int s = tid; s < nslots; s += 256) {
        int lane = s & 31;
        int n    = (s >> 5) & 7;
        int ks   = s >> 8;
        int h    = lane >> 4;
        int l    = lane & 15;
        int krow = ks * 4 + 2 * h;
        int col  = n * 16 + l;
        unsigned ldsb = (sw_float_off + (unsigned)((ks * 8 + n) * 64 + lane * 2)) * 4u;
        const float* g0 = W + (size_t)krow * 128 + col;
        async_g2l_b32(ldsb,      g0);          // W[krow  ][col]
        async_g2l_b32(ldsb + 4u, g0 + 128);    // W[krow+1][col]
    }
}

__device__ __forceinline__ v2f frag(const float* __restrict__ sw, int ks, int n, int lane) {
    return *(const v2f*)(sw + (ks * 8 + n) * 64 + lane * 2);   // ds_load_b64 / 2addr
}

// One K-step: load all 8 B fragments first (clause-able DS loads, incremental
// DScnt drain), then the 8-WMMA octet against a shared A fragment.
#define KSTEP_WMMA(ACC, SW, KS)                                                  \
    {                                                                            \
        v2f bw[8];                                                               \
        _Pragma("unroll")                                                        \
        for (int n = 0; n < 8; ++n) bw[n] = frag((SW), (KS), n, lane);           \
        _Pragma("unroll")                                                        \
        for (int n = 0; n < 8; ++n)                                              \
            ACC[n] = __builtin_amdgcn_wmma_f32_16x16x4_f32(                      \
                false, a, false, bw[n], (short)0, ACC[n], false, false);         \
    }

__global__ void k_zero(float* __restrict__ p, size_t n) {
    size_t i = (size_t)blockIdx.x * blockDim.x + threadIdx.x;
    size_t stride = (size_t)gridDim.x * blockDim.x;
    for (; i < n; i += stride) p[i] = 0.0f;
}

// ---------------------------------------------------------------------------
// v = x @ lin1_w + lin1_b        persistent, LDS-resident swizzled weights
// LDS: [0, 16384) floats = lin1_w fragments (64 KB)
// ---------------------------------------------------------------------------
__global__ __launch_bounds__(256) void k_node_proj(
    const float* __restrict__ x, const float* __restrict__ w,
    const float* __restrict__ b, float* __restrict__ v, int tiles)
{
    extern __shared__ float smem[];
    const int tid  = threadIdx.x;
    const int lane = tid & 31;
    const int wave = tid >> 5;
    const int l    = lane & 15;
    const int half = lane >> 4;

    fill_sw(0u, w, 32, tid);
    async_drain();
    __syncthreads();
    const float* sw = smem;

    const v8f vzero = {0.f,0.f,0.f,0.f,0.f,0.f,0.f,0.f};

    for (int t = blockIdx.x * 8 + wave; t < tiles; t += gridDim.x * 8) {
        const int rowBase = t * 16;
        v8f acc[8];
#pragma unroll
        for (int n = 0; n < 8; ++n) acc[n] = vzero;

        const float* arow = x + (size_t)(rowBase + l) * 128;
        for (int ks = 0; ks < 32; ++ks) {
            v2f a = *(const v2f*)(arow + ks * 4 + 2 * half);
            KSTEP_WMMA(acc, sw, ks)
        }
#pragma unroll
        for (int n = 0; n < 8; ++n) {
            float bias = b[n * 16 + l];
#pragma unroll
            for (int g = 0; g < 8; ++g) {
                int row = rowBase + g + 8 * half;
                v[(size_t)row * 128 + n * 16 + l] = acc[n][g] + bias;
            }
        }
    }
}

// ---------------------------------------------------------------------------
// Fused edge pipeline (persistent):
//   h1 = ssp(edge_attr @ cf1_w); h2 = ssp(h1 @ cf2_w + cf2_b)
//   atomicAdd(agg[dst], v[src] * h2 * ew)
// LDS floats: cf1 swz [0,8192) | cf2 swz [8192,24576) | hbuf [24576,+8*2112)
// ---------------------------------------------------------------------------
__global__ __launch_bounds__(256) void k_edge(
    const float* __restrict__ edge_attr,   // [E,64]
    const float* __restrict__ ew,          // [E]
    const float* __restrict__ cf1_w,       // [64,128]
    const float* __restrict__ cf2_w,       // [128,128]
    const float* __restrict__ cf2_b,       // [128]
    const float* __restrict__ vtab,        // [N,128]
    const int*   __restrict__ src,
    const int*   __restrict__ dst,
    float* __restrict__ agg, int tiles)
{
    extern __shared__ float smem[];
    const int tid  = threadIdx.x;
    const int lane = tid & 31;
    const int wave = tid >> 5;
    const int l    = lane & 15;
    const int half = lane >> 4;

    fill_sw(0u,    cf1_w, 16, tid);
    fill_sw(8192u, cf2_w, 32, tid);
    async_drain();
    __syncthreads();
    const float* sw1 = smem;
    const float* sw2 = smem + 8192;
    float* hb = smem + 24576 + wave * (16 * HSTRIDE);

    const v8f vzero = {0.f,0.f,0.f,0.f,0.f,0.f,0.f,0.f};
    float bias[8];
#pragma unroll
    for (int n = 0; n < 8; ++n) bias[n] = cf2_b[n * 16 + l];

    for (int t = blockIdx.x * 8 + wave; t < tiles; t += gridDim.x * 8) {
        const int eBase = t * 16;

        // ---- stage 1: h1 = edge_attr @ cf1_w -------------------------------
        v8f acc[8];
#pragma unroll
        for (int n = 0; n < 8; ++n) acc[n] = vzero;

        const float* arow = edge_attr + (size_t)(eBase + l) * 64;
        for (int ks = 0; ks < 16; ++ks) {
            v2f a = *(const v2f*)(arow + ks * 4 + 2 * half);
            KSTEP_WMMA(acc, sw1, ks)
        }

        // ssp -> per-wave LDS tile (C-layout -> row-major, padded stride)
#pragma unroll
        for (int n = 0; n < 8; ++n) {
#pragma unroll
            for (int g = 0; g < 8; ++g) {
                int r = g + 8 * half;
                hb[r * HSTRIDE + n * 16 + l] = ssp_f(acc[n][g]);
            }
        }
        asm volatile("s_wait_dscnt 0" ::: "memory");   // wave-local LDS RAW

        // ---- stage 2: h2 = h1 @ cf2_w --------------------------------------
        v8f acc2[8];
#pragma unroll
        for (int n = 0; n < 8; ++n) acc2[n] = vzero;

        for (int ks = 0; ks < 32; ++ks) {
            v2f a = *(const v2f*)(hb + l * HSTRIDE + ks * 4 + 2 * half);
            KSTEP_WMMA(acc2, sw2, ks)
        }

        // ---- stage 3: bias + ssp + gather v[src] + modulate + scatter ------
#pragma unroll
        for (int g = 0; g < 8; ++g) {
            int e = eBase + g + 8 * half;
            int s = src[e];
            int d = dst[e];
            float wgt = ew[e];
            const float* vrow = vtab + (size_t)s * 128;
            float* orow = agg + (size_t)d * 128;
#pragma unroll
            for (int n = 0; n < 8; ++n) {
                int col = n * 16 + l;
                float m = ssp_f(acc2[n][g] + bias[n]);
                m = m * vrow[col] * wgt;
                unsafeAtomicAdd(&orow[col], m);   // global_atomic_add_f32 (no return)
            }
        }
    }
}

// ---------------------------------------------------------------------------
// out = x + (ssp(agg @ lin2_w + b2) @ lin3_w + b3)     persistent
// LDS floats: w2 swz [0,16384) | w3 swz [16384,32768) | tbuf [32768,+8*2112)
// ---------------------------------------------------------------------------
__global__ __launch_bounds__(256) void k_post(
    const float* __restrict__ agg,
    const float* __restrict__ w2, const float* __restrict__ b2,
    const float* __restrict__ w3, const float* __restrict__ b3,
    const float* __restrict__ x, float* __restrict__ out, int tiles)
{
    extern __shared__ float smem[];
    const int tid  = threadIdx.x;
    const int lane = tid & 31;
    const int wave = tid >> 5;
    const int l    = lane & 15;
    const int half = lane >> 4;

    fill_sw(0u,     w2, 32, tid);
    fill_sw(16384u, w3, 32, tid);
    async_drain();
    __syncthreads();
    const float* sw2 = smem;
    const float* sw3 = smem + 16384;
    float* tb = smem + 32768 + wave * (16 * HSTRIDE);

    const v8f vzero = {0.f,0.f,0.f,0.f,0.f,0.f,0.f,0.f};

    for (int t = blockIdx.x * 8 + wave; t < tiles; t += gridDim.x * 8) {
        const int rowBase = t * 16;

        v8f acc[8];
#pragma unroll
        for (int n = 0; n < 8; ++n) acc[n] = vzero;

        const float* arow = agg + (size_t)(rowBase + l) * 128;
        for (int ks = 0; ks < 32; ++ks) {
            v2f a = *(const v2f*)(arow + ks * 4 + 2 * half);
            KSTEP_WMMA(acc, sw2, ks)
        }

#pragma unroll
        for (int n = 0; n < 8; ++n) {
            float bias = b2[n * 16 + l];
#pragma unroll
            for (int g = 0; g < 8; ++g) {
                int r = g + 8 * half;
                tb[r * HSTRIDE + n * 16 + l] = ssp_f(acc[n][g] + bias);
            }
        }
        asm volatile("s_wait_dscnt 0" ::: "memory");

        v8f acc2[8];
#pragma unroll
        for (int n = 0; n < 8; ++n) acc2[n] = vzero;

        for (int ks = 0; ks < 32; ++ks) {
            v2f a = *(const v2f*)(tb + l * HSTRIDE + ks * 4 + 2 * half);
            KSTEP_WMMA(acc2, sw3, ks)
        }

#pragma unroll
        for (int n = 0; n < 8; ++n) {
            float bias = b3[n * 16 + l];
#pragma unroll
            for (int g = 0; g < 8; ++g) {
                int row = rowBase + g + 8 * half;
                size_t idx = (size_t)row * 128 + n * 16 + l;
                out[idx] = x[idx] + acc2[n][g] + bias;
            }
        }
    }
}

// ---------------------------------------------------------------------------
extern "C" void kernel_launch(void* const* d_in, const int* in_sizes, int n_in,
                              void* d_out, int out_size, void* d_ws, size_t ws_size,
                              hipStream_t stream) {
    const float* x           = (const float*)d_in[0];
    const float* edge_attr   = (const float*)d_in[1];
    const float* edge_weight = (const float*)d_in[2];
    const float* lin1_w      = (const float*)d_in[3];
    const float* lin1_b      = (const float*)d_in[4];
    const float* cf1_w       = (const float*)d_in[5];
    const float* cf2_w       = (const float*)d_in[6];
    const float* cf2_b       = (const float*)d_in[7];
    const float* lin2_w      = (const float*)d_in[8];
    const float* lin2_b      = (const float*)d_in[9];
    const float* lin3_w      = (const float*)d_in[10];
    const float* lin3_b      = (const float*)d_in[11];
    const int*   edge_index  = (const int*)d_in[12];

    const int N = in_sizes[0] / 128;   // 100000
    const int E = in_sizes[1] / 64;    // 1600000
    const int* src = edge_index;       // row 0
    const int* dst = edge_index + E;   // row 1

    float* v   = (float*)d_ws;
    float* agg = (float*)((char*)d_ws + (size_t)N * 128 * sizeof(float));
    float* out = (float*)d_out;

    const int nodeTiles = (N + 15) / 16;   // 6250
    const int edgeTiles = (E + 15) / 16;   // 100000

    const size_t smemNode = 16384u * 4u;                               //  64 KB
    const size_t smemEdge = (24576u + 8u * 16u * HSTRIDE) * 4u;        // ~162 KB
    const size_t smemPost = (32768u + 8u * 16u * HSTRIDE) * 4u;        // ~194 KB

    (void)hipFuncSetAttribute(reinterpret_cast<const void*>(k_node_proj),
                              hipFuncAttributeMaxDynamicSharedMemorySize, (int)smemNode);
    (void)hipFuncSetAttribute(reinterpret_cast<const void*>(k_edge),
                              hipFuncAttributeMaxDynamicSharedMemorySize, (int)smemEdge);
    (void)hipFuncSetAttribute(reinterpret_cast<const void*>(k_post),
                              hipFuncAttributeMaxDynamicSharedMemorySize, (int)smemPost);

    // agg must start at zero every call (harness poisons ws)
    k_zero<<<2048, 256, 0, stream>>>(agg, (size_t)N * 128);

    k_node_proj<<<512, 256, smemNode, stream>>>(x, lin1_w, lin1_b, v, nodeTiles);

    k_edge<<<2048, 256, smemEdge, stream>>>(edge_attr, edge_weight, cf1_w, cf2_w,
                                            cf2_b, v, src, dst, agg, edgeTiles);

    k_post<<<512, 256, smemPost, stream>>>(agg, lin2_w, lin2_b, lin3_w, lin3_b,
                                           x, out, nodeTiles);
}